// Izhikevich_SNN_34342558499577
// MI455X (gfx1250) — compile-verified
//
#include <hip/hip_runtime.h>

// Izhikevich SNN scan for MI455X (gfx1250).
//
// Shape: x[seq=2048][n=16384], state[3][n], weight[1].
// Outputs (concatenated flat in d_out):
//   outputs[seq][n]        (spikes)
//   states [seq][3][n]     (stacked v_r, u_r, s per timestep)
//
// Memory-bound: 128 MiB read + 512 MiB written => ~27us at 23.3 TB/s.
// Strategy: one thread per neuron (recurrence is sequential in t),
// time-unrolled x4 for MLP, nontemporal loads/stores (streamed once,
// larger than L2), and global_prefetch of the x stream ~12 rows ahead
// to compensate for the low wave count (512 wave32s).

namespace {
constexpr float P_A      = 0.01f;
constexpr float P_B      = 0.2f;
constexpr float P_C      = -65.0f;
constexpr float P_D      = 8.0f;
constexpr float P_THRESH = 30.0f;
constexpr float P_DT     = 0.25f;
constexpr int   UNROLL   = 4;
constexpr int   PF_DIST  = 12;   // prefetch distance in timesteps (rows of x)
}

__global__ __launch_bounds__(128)
void izh_scan_kernel(const float* __restrict__ x,
                     const float* __restrict__ state,
                     const float* __restrict__ weight,
                     float* __restrict__ out_spk,   // [seq][n]
                     float* __restrict__ out_sta,   // [seq][3][n]
                     int seq, int n)
{
    const int tid = blockIdx.x * blockDim.x + threadIdx.x;
    if (tid >= n) return;

    const size_t N = (size_t)n;

    // Carry state: v, u (s is not carried; recomputed each step).
    float v = state[tid];
    float u = state[N + tid];
    const float w = weight[0];          // uniform -> scalar load

    const float* xp  = x       + tid;
    float*       spk = out_spk + tid;
    float*       sta = out_sta + tid;

    int t = 0;
    for (; t + UNROLL <= seq; t += UNROLL) {
        // Stream in UNROLL timesteps of input (coalesced across lanes,
        // rows are 64 KB apart). NT: consumed exactly once.
        float xi[UNROLL];
#pragma unroll
        for (int k = 0; k < UNROLL; ++k)
            xi[k] = __builtin_nontemporal_load(xp + (size_t)(t + k) * N);

        // Prefetch the x stream ahead into the cache hierarchy
        // (global_prefetch_b8) to keep HBM busy despite only 512 waves.
        if (t + PF_DIST + UNROLL <= seq) {
#pragma unroll
            for (int k = 0; k < UNROLL; ++k)
                __builtin_prefetch(xp + (size_t)(t + PF_DIST + k) * N, 0, 0);
        }

#pragma unroll
        for (int k = 0; k < UNROLL; ++k) {
            const float I   = w * xi[k];
            // Euler membrane update (uses old u), then recovery (uses old v).
            const float v_new = v + P_DT * (0.04f * v * v + 5.0f * v + 140.0f - u + I);
            const float u_new = u + P_DT * P_A * (P_B * v - u);
            // Heaviside spike on (v_new - thresh).
            const float s   = (v_new - P_THRESH >= 0.0f) ? 1.0f : 0.0f;
            const float v_r = v_new * (1.0f - s) + P_C * s;
            const float u_r = u_new + P_D * s;
            v = v_r;
            u = u_r;

            // 16 B of output per element, all streamed once -> NT stores.
            const size_t r = (size_t)(t + k);
            __builtin_nontemporal_store(s,   spk + r * N);
            float* srow = sta + r * 3u * N;
            __builtin_nontemporal_store(v_r, srow);
            __builtin_nontemporal_store(u_r, srow + N);
            __builtin_nontemporal_store(s,   srow + 2u * N);
        }
    }

    // Remainder (seq=2048 is divisible by 4; kept for generality).
    for (; t < seq; ++t) {
        const float xi  = __builtin_nontemporal_load(xp + (size_t)t * N);
        const float I   = w * xi;
        const float v_new = v + P_DT * (0.04f * v * v + 5.0f * v + 140.0f - u + I);
        const float u_new = u + P_DT * P_A * (P_B * v - u);
        const float s   = (v_new - P_THRESH >= 0.0f) ? 1.0f : 0.0f;
        const float v_r = v_new * (1.0f - s) + P_C * s;
        const float u_r = u_new + P_D * s;
        v = v_r;
        u = u_r;
        const size_t r = (size_t)t;
        __builtin_nontemporal_store(s,   spk + r * N);
        float* srow = sta + r * 3u * N;
        __builtin_nontemporal_store(v_r, srow);
        __builtin_nontemporal_store(u_r, srow + N);
        __builtin_nontemporal_store(s,   srow + 2u * N);
    }
}

extern "C" void kernel_launch(void* const* d_in, const int* in_sizes, int n_in,
                              void* d_out, int out_size, void* d_ws, size_t ws_size,
                              hipStream_t stream) {
    (void)n_in; (void)d_ws; (void)ws_size;

    const float* x      = (const float*)d_in[0];   // [seq][n]
    const float* state  = (const float*)d_in[1];   // [3][n]
    const float* weight = (const float*)d_in[2];   // [1]

    const int n   = in_sizes[1] / 3;               // 16384
    const int seq = in_sizes[0] / n;               // 2048
    (void)out_size;                                // == seq*n + seq*3*n

    float* out_spk = (float*)d_out;
    float* out_sta = out_spk + (size_t)seq * (size_t)n;

    const int block = 128;                         // 4 wave32s per block
    const int grid  = (n + block - 1) / block;     // 128 blocks -> wide WGP spread

    izh_scan_kernel<<<grid, block, 0, stream>>>(x, state, weight,
                                                out_spk, out_sta, seq, n);
}